// VariationalGCNEncoder_68186900791376
// MI455X (gfx1250) — compile-verified
//
#include <hip/hip_runtime.h>
#include <hip/hip_bf16.h>

typedef __attribute__((ext_vector_type(2))) float v2f;
typedef __attribute__((ext_vector_type(8))) float v8f;

#define NN   100000
#define EE   3200000
#define CIN  256
#define CHID 128
#define COUT 64
#define NSTRIPS (NN / 16)   // 6250, exact

// ---------------- degree / normalization ----------------

__global__ void k_fill1(float* deg) {
    int i = blockIdx.x * blockDim.x + threadIdx.x;
    if (i < NN) deg[i] = 1.0f;   // self-loop contribution
}

__global__ void k_deg(const int* __restrict__ dst, float* __restrict__ deg) {
    int e = blockIdx.x * blockDim.x + threadIdx.x;
    if (e < EE) atomicAdd(&deg[dst[e]], 1.0f);
}

__global__ void k_dinv(float* deg) {
    int i = blockIdx.x * blockDim.x + threadIdx.x;
    if (i < NN) {
        float d = deg[i];
        deg[i] = d > 0.0f ? rsqrtf(d) : 0.0f;
    }
}

// ---------------- GEMM 1: H_pre[N,128] = X[N,256] @ W1[256,128] (fp32 WMMA) ----------------
// One wave per 16-row strip; A fragment reused across all 8 column tiles.

__global__ void k_gemm1(const float* __restrict__ X, const float* __restrict__ W,
                        float* __restrict__ H) {
    int warp  = threadIdx.x >> 5;
    int strip = blockIdx.x * 8 + warp;
    if (strip >= NSTRIPS) return;          // wave-uniform: EXEC stays all-1s
    int lane = threadIdx.x & 31;
    int half = lane >> 4;                  // 0: K pair {k,k+1}, 1: {k+2,k+3}
    int l    = lane & 15;
    int m0   = strip * 16;

    v8f acc[8] = {};
    const float* ap0 = X + (size_t)(m0 + l) * CIN;

    for (int k = 0; k < CIN; k += 4) {
        v2f a;
        a.x = ap0[k + 2 * half];
        a.y = ap0[k + 2 * half + 1];
#pragma unroll
        for (int t = 0; t < 8; ++t) {
            const float* bp = W + (size_t)(k + 2 * half) * CHID + t * 16 + l;
            v2f b;
            b.x = bp[0];
            b.y = bp[CHID];
            acc[t] = __builtin_amdgcn_wmma_f32_16x16x4_f32(
                false, a, false, b, (short)0, acc[t], false, false);
        }
    }
#pragma unroll
    for (int t = 0; t < 8; ++t)
#pragma unroll
        for (int r = 0; r < 8; ++r)
            H[(size_t)(m0 + r + 8 * half) * CHID + t * 16 + l] = acc[t][r];
}

// ---------------- GEMM 2+3 fused: Tmu = H @ Wmu, Tls = H @ Wls (both [128,64]) ----------------

__global__ void k_gemm23(const float* __restrict__ H, const float* __restrict__ Wmu,
                         const float* __restrict__ Wls, float* __restrict__ Tmu,
                         float* __restrict__ Tls) {
    int warp  = threadIdx.x >> 5;
    int strip = blockIdx.x * 8 + warp;
    if (strip >= NSTRIPS) return;
    int lane = threadIdx.x & 31;
    int half = lane >> 4;
    int l    = lane & 15;
    int m0   = strip * 16;

    v8f am[4] = {};
    v8f al[4] = {};
    const float* ap0 = H + (size_t)(m0 + l) * CHID;

    for (int k = 0; k < CHID; k += 4) {
        v2f a;
        a.x = ap0[k + 2 * half];
        a.y = ap0[k + 2 * half + 1];
#pragma unroll
        for (int t = 0; t < 4; ++t) {
            const float* bp = Wmu + (size_t)(k + 2 * half) * COUT + t * 16 + l;
            v2f b;
            b.x = bp[0];
            b.y = bp[COUT];
            am[t] = __builtin_amdgcn_wmma_f32_16x16x4_f32(
                false, a, false, b, (short)0, am[t], false, false);
        }
#pragma unroll
        for (int t = 0; t < 4; ++t) {
            const float* bp = Wls + (size_t)(k + 2 * half) * COUT + t * 16 + l;
            v2f b;
            b.x = bp[0];
            b.y = bp[COUT];
            al[t] = __builtin_amdgcn_wmma_f32_16x16x4_f32(
                false, a, false, b, (short)0, al[t], false, false);
        }
    }
#pragma unroll
    for (int t = 0; t < 4; ++t)
#pragma unroll
        for (int r = 0; r < 8; ++r) {
            size_t row = (size_t)(m0 + r + 8 * half);
            Tmu[row * COUT + t * 16 + l] = am[t][r];
            Tls[row * COUT + t * 16 + l] = al[t][r];
        }
}

// ---------------- bias init ----------------

__global__ void k_init_h(float* __restrict__ h, const float* __restrict__ b1) {
    long i = (long)blockIdx.x * blockDim.x + threadIdx.x;
    if (i < (long)NN * CHID) h[i] = b1[i & (CHID - 1)];
}

__global__ void k_init_out(float* __restrict__ out, const float* __restrict__ bmu,
                           const float* __restrict__ bls) {
    long i = (long)blockIdx.x * blockDim.x + threadIdx.x;
    const long halfsz = (long)NN * COUT;
    if (i < halfsz)            out[i] = bmu[i & (COUT - 1)];
    else if (i < 2 * halfsz)   out[i] = bls[i & (COUT - 1)];
}

// ---------------- message passing (scatter-add), one wave32 per edge ----------------

__global__ void k_scatter1(const int* __restrict__ src, const int* __restrict__ dst,
                           const float* __restrict__ dinv, const float* __restrict__ Hin,
                           float* __restrict__ Hout) {
    int e    = (int)(((long)blockIdx.x * blockDim.x + threadIdx.x) >> 5);
    int lane = threadIdx.x & 31;
    if (e >= EE + NN) return;
    int s, d;
    if (e < EE) { s = src[e]; d = dst[e]; }
    else        { s = d = e - EE; }            // self-loop
    float nrm = dinv[s] * dinv[d];
    const float4 v = *(const float4*)(Hin + (size_t)s * CHID + lane * 4);
    float* o = Hout + (size_t)d * CHID + lane * 4;
    atomicAdd(o + 0, v.x * nrm);
    atomicAdd(o + 1, v.y * nrm);
    atomicAdd(o + 2, v.z * nrm);
    atomicAdd(o + 3, v.w * nrm);
}

__global__ void k_scatter23(const int* __restrict__ src, const int* __restrict__ dst,
                            const float* __restrict__ dinv, const float* __restrict__ Tmu,
                            const float* __restrict__ Tls, float* __restrict__ Omu,
                            float* __restrict__ Ols) {
    int e    = (int)(((long)blockIdx.x * blockDim.x + threadIdx.x) >> 5);
    int lane = threadIdx.x & 31;
    if (e >= EE + NN) return;
    int s, d;
    if (e < EE) { s = src[e]; d = dst[e]; }
    else        { s = d = e - EE; }
    float nrm = dinv[s] * dinv[d];
    const float2 m = *(const float2*)(Tmu + (size_t)s * COUT + lane * 2);
    const float2 g = *(const float2*)(Tls + (size_t)s * COUT + lane * 2);
    float* om = Omu + (size_t)d * COUT + lane * 2;
    float* ol = Ols + (size_t)d * COUT + lane * 2;
    atomicAdd(om + 0, m.x * nrm);
    atomicAdd(om + 1, m.y * nrm);
    atomicAdd(ol + 0, g.x * nrm);
    atomicAdd(ol + 1, g.y * nrm);
}

__global__ void k_relu(float* __restrict__ h) {
    long i = (long)blockIdx.x * blockDim.x + threadIdx.x;
    if (i < (long)NN * CHID) h[i] = fmaxf(h[i], 0.0f);
}

// ---------------- launch ----------------

extern "C" void kernel_launch(void* const* d_in, const int* in_sizes, int n_in,
                              void* d_out, int out_size, void* d_ws, size_t ws_size,
                              hipStream_t stream) {
    (void)in_sizes; (void)n_in; (void)out_size; (void)ws_size;

    const float* x    = (const float*)d_in[0];
    const int*   ei   = (const int*)d_in[1];      // [2,E] flattened: src then dst
    const float* W1   = (const float*)d_in[2];
    const float* b1   = (const float*)d_in[3];
    const float* Wmu  = (const float*)d_in[4];
    const float* bmu  = (const float*)d_in[5];
    const float* Wls  = (const float*)d_in[6];
    const float* bls  = (const float*)d_in[7];

    const int* src = ei;
    const int* dst = ei + EE;

    float* out = (float*)d_out;                   // [mu (N*64) | logstd (N*64)]

    // workspace: dinv | hpre (reused as tmu/tls) | h
    float* ws   = (float*)d_ws;
    float* dinv = ws;                               // N
    float* hpre = ws + NN;                          // N*CHID
    float* h    = hpre + (size_t)NN * CHID;         // N*CHID
    float* tmu  = hpre;                             // N*COUT (reuse)
    float* tls  = hpre + (size_t)NN * COUT;         // N*COUT (reuse)

    const int T = 256;
    const unsigned gN    = (NN + T - 1) / T;
    const unsigned gE    = (EE + T - 1) / T;
    const unsigned gElem = ((unsigned)((long)NN * CHID) + T - 1) / T;   // 50000
    const unsigned gOut  = ((unsigned)(2L * NN * COUT) + T - 1) / T;    // 50000
    const unsigned gGemm = (NSTRIPS + 7) / 8;                            // 782
    const unsigned gEdge = (EE + NN + 7) / 8;                            // 412500 (wave/edge)

    // 1) normalization
    k_fill1<<<gN, T, 0, stream>>>(dinv);
    k_deg  <<<gE, T, 0, stream>>>(dst, dinv);
    k_dinv <<<gN, T, 0, stream>>>(dinv);

    // 2) conv1: dense transform -> scatter-add -> relu
    k_gemm1   <<<gGemm, T, 0, stream>>>(x, W1, hpre);
    k_init_h  <<<gElem, T, 0, stream>>>(h, b1);
    k_scatter1<<<gEdge, T, 0, stream>>>(src, dst, dinv, hpre, h);
    k_relu    <<<gElem, T, 0, stream>>>(h);

    // 3) conv2 + conv3 fused (mu, logstd)
    k_gemm23   <<<gGemm, T, 0, stream>>>(h, Wmu, Wls, tmu, tls);
    k_init_out <<<gOut, T, 0, stream>>>(out, bmu, bls);
    k_scatter23<<<gEdge, T, 0, stream>>>(src, dst, dinv, tmu, tls, out, out + (size_t)NN * COUT);
}